// LightGCN_39213051412907
// MI455X (gfx1250) — compile-verified
//
#include <hip/hip_runtime.h>

#define N_USERS 60000
#define N_ITEMS 30000
#define N_NODE  (N_USERS + N_ITEMS)
#define N_EDGE  2000000
#define EMB     128
#define BATCH   16384

typedef float v2f __attribute__((ext_vector_type(2)));
typedef float v8f __attribute__((ext_vector_type(8)));

// ---------------------------------------------------------------------------
// init: acc = xa = ego = cat(user_emb, item_emb);  xb = 0
// ---------------------------------------------------------------------------
__global__ __launch_bounds__(256) void lgcn_init(
    const float* __restrict__ ue, const float* __restrict__ ie,
    float* __restrict__ acc, float* __restrict__ xa, float* __restrict__ xb) {
  long long i = ((long long)blockIdx.x * blockDim.x + threadIdx.x) * 4;
  const long long total = (long long)N_NODE * EMB;
  if (i >= total) return;
  const long long ulim = (long long)N_USERS * EMB;
  float4 v = (i < ulim) ? *(const float4*)(ue + i)
                        : *(const float4*)(ie + (i - ulim));
  *(float4*)(acc + i) = v;
  *(float4*)(xa + i)  = v;
  *(float4*)(xb + i)  = make_float4(0.f, 0.f, 0.f, 0.f);
}

// ---------------------------------------------------------------------------
// SpMM: y[row] += val * x[col]   (one wave32 per edge, float4 per lane)
// Edge streams read non-temporally so the 46MB node matrix stays L2-resident.
// ---------------------------------------------------------------------------
__global__ __launch_bounds__(256) void lgcn_spmm(
    const int* __restrict__ rows, const int* __restrict__ cols,
    const float* __restrict__ vals,
    const float* __restrict__ x, float* __restrict__ y) {
  const int e    = (int)(((long long)blockIdx.x * blockDim.x + threadIdx.x) >> 5);
  const int lane = threadIdx.x & 31;
  if (e >= N_EDGE) return;
  const int   r = __builtin_nontemporal_load(rows + e);
  const int   c = __builtin_nontemporal_load(cols + e);
  const float v = __builtin_nontemporal_load(vals + e);
  const float4 xv = *(const float4*)(x + (long long)c * EMB + lane * 4);
  float* yb = y + (long long)r * EMB + lane * 4;
  unsafeAtomicAdd(yb + 0, v * xv.x);
  unsafeAtomicAdd(yb + 1, v * xv.y);
  unsafeAtomicAdd(yb + 2, v * xv.z);
  unsafeAtomicAdd(yb + 3, v * xv.w);
}

// ---------------------------------------------------------------------------
// acc += src; optionally zero tgt (the next SpMM destination)
// ---------------------------------------------------------------------------
__global__ __launch_bounds__(256) void lgcn_accum(
    float* __restrict__ acc, const float* __restrict__ src,
    float* __restrict__ tgt, int zero_tgt) {
  long long i = ((long long)blockIdx.x * blockDim.x + threadIdx.x) * 4;
  if (i >= (long long)N_NODE * EMB) return;
  float4 a = *(float4*)(acc + i);
  float4 s = *(const float4*)(src + i);
  a.x += s.x; a.y += s.y; a.z += s.z; a.w += s.w;
  *(float4*)(acc + i) = a;
  if (zero_tgt) *(float4*)(tgt + i) = make_float4(0.f, 0.f, 0.f, 0.f);
}

// ---------------------------------------------------------------------------
// Scoring: per tile of 16 pairs, D = U x I^T via V_WMMA_F32_16X16X4_F32,
// take the diagonal, scale by 1/16 (the /(L+1) on each side).
// One wave per tile; 2 waves / 64-thread block; LDS row pitch 132 floats
// so the stride-132 float2 fragment reads hit distinct banks.
// ---------------------------------------------------------------------------
__global__ __launch_bounds__(64) void lgcn_score(
    const float* __restrict__ acc, const int* __restrict__ users,
    const int* __restrict__ items, float* __restrict__ out) {
  __shared__ float sm[2][2][16][132];
  const int wave = threadIdx.x >> 5;
  const int lane = threadIdx.x & 31;
  const int tile = blockIdx.x * 2 + wave;
  const int base = tile * 16;

  float (*sU)[132] = sm[wave][0];
  float (*sI)[132] = sm[wave][1];

  // Stage 16 user rows + 16 item rows; whole wave loads one row (b128/lane).
  for (int m = 0; m < 16; ++m) {
    const int u  = users[base + m];
    const int it = items[base + m];
    const float4 ru = *(const float4*)(acc + (long long)u * EMB + lane * 4);
    const float4 ri = *(const float4*)(acc + (long long)(N_USERS + it) * EMB + lane * 4);
    *(float4*)&sU[m][lane * 4] = ru;
    *(float4*)&sI[m][lane * 4] = ri;
  }
  __syncthreads();

  // A[m][k] = U_row_m[k];  B[k][n] = I_row_n[k]  -> D[b][b] = dot(U_b, I_b).
  // 32-bit A 16x4 lane layout: lane L holds row L%16, K {0,1} (L<16) or {2,3}.
  // B 4x16 mirrors it with N = L%16 -> identical (row, k) addressing.
  const int row  = lane & 15;
  const int khi  = (lane >> 4) << 1;  // 0 or 2
  v8f c = {0.f, 0.f, 0.f, 0.f, 0.f, 0.f, 0.f, 0.f};
#pragma unroll
  for (int kk = 0; kk < EMB / 4; ++kk) {
    const int k0 = kk * 4 + khi;
    v2f a, b;
    a.x = sU[row][k0];  a.y = sU[row][k0 + 1];
    b.x = sI[row][k0];  b.y = sI[row][k0 + 1];
    c = __builtin_amdgcn_wmma_f32_16x16x4_f32(
        /*neg_a=*/false, a, /*neg_b=*/false, b,
        /*c_mod=*/(short)0, c, /*reuse_a=*/false, /*reuse_b=*/false);
  }

  // Diagonal of D: VGPR r, lane r (m=0..7) and VGPR r, lane r+24 (m=8..15).
  const int  L      = lane;
  const bool active = (L < 8) || (L >= 24);
  const int  r      = L & 7;
  float d = c[0];
  if (r == 1) d = c[1];
  if (r == 2) d = c[2];
  if (r == 3) d = c[3];
  if (r == 4) d = c[4];
  if (r == 5) d = c[5];
  if (r == 6) d = c[6];
  if (r == 7) d = c[7];
  if (active) out[base + ((L < 8) ? L : (L - 16))] = d * 0.0625f;
}

// ---------------------------------------------------------------------------
extern "C" void kernel_launch(void* const* d_in, const int* in_sizes, int n_in,
                              void* d_out, int out_size, void* d_ws, size_t ws_size,
                              hipStream_t stream) {
  const float* ue    = (const float*)d_in[0];
  const float* ie    = (const float*)d_in[1];
  const int*   arow  = (const int*)  d_in[2];
  const int*   acol  = (const int*)  d_in[3];
  const float* aval  = (const float*)d_in[4];
  const int*   users = (const int*)  d_in[5];
  const int*   items = (const int*)  d_in[6];
  float*       out   = (float*)d_out;

  const size_t NFE = (size_t)N_NODE * EMB;
  float* acc = (float*)d_ws;       // 46.1 MB
  float* xa  = acc + NFE;          // 46.1 MB
  float* xb  = xa + NFE;           // 46.1 MB

  const int elem4      = (int)(NFE / 4);
  const int initBlocks = (elem4 + 255) / 256;
  const int spmmBlocks = (int)(((long long)N_EDGE * 32 + 255) / 256);

  lgcn_init<<<initBlocks, 256, 0, stream>>>(ue, ie, acc, xa, xb);

  // layer 1: xa -> xb
  lgcn_spmm<<<spmmBlocks, 256, 0, stream>>>(arow, acol, aval, xa, xb);
  lgcn_accum<<<initBlocks, 256, 0, stream>>>(acc, xb, xa, 1);
  // layer 2: xb -> xa
  lgcn_spmm<<<spmmBlocks, 256, 0, stream>>>(arow, acol, aval, xb, xa);
  lgcn_accum<<<initBlocks, 256, 0, stream>>>(acc, xa, xb, 1);
  // layer 3: xa -> xb
  lgcn_spmm<<<spmmBlocks, 256, 0, stream>>>(arow, acol, aval, xa, xb);
  lgcn_accum<<<initBlocks, 256, 0, stream>>>(acc, xb, nullptr, 0);

  // scoring: 1024 tiles of 16 pairs, 2 waves per block
  lgcn_score<<<BATCH / 32, 64, 0, stream>>>(acc, users, items, out);
}